// RecurrentGraphNeuralNet_43361989821071
// MI455X (gfx1250) — compile-verified
//
#include <hip/hip_runtime.h>
#include <stdint.h>

typedef __attribute__((ext_vector_type(2))) float v2f;
typedef __attribute__((ext_vector_type(8))) float v8f;

#define N_NODES 100000
#define N_EDGES 1600000
#define HID     128
#define PRED_C  64
#define NTILES  (N_NODES / 16)   // 6250, exact

__device__ __forceinline__ v2f ld2(const float* p) {
    return *(const v2f*)p;
}

// ---------------------------------------------------------------------------
// Kernel 0: zero the agg accumulator (must be re-zeroed every launch).
// ---------------------------------------------------------------------------
__global__ __launch_bounds__(256) void gnn_zero_kernel(float4* __restrict__ agg, int n4) {
    int i = blockIdx.x * 256 + threadIdx.x;
    if (i < n4) agg[i] = make_float4(0.f, 0.f, 0.f, 0.f);
}

// ---------------------------------------------------------------------------
// Kernel 1: h = x @ W_w^T + W_b   (h[n][j] = sum_k x[n][k]*W_w[j][k] + W_b[j])
// One wave per 16-node tile; wave computes the full 16x128 output panel with
// v_wmma_f32_16x16x4_f32 (32 k-steps x 8 column tiles = 256 WMMAs).
// ---------------------------------------------------------------------------
__global__ __launch_bounds__(256) void gnn_gemm_h_kernel(const float* __restrict__ x,
                                                         const float* __restrict__ Ww,
                                                         const float* __restrict__ Wb,
                                                         float* __restrict__ h) {
    const int lane  = threadIdx.x & 31;
    const int wave  = threadIdx.x >> 5;
    const int tile  = blockIdx.x * 8 + wave;
    if (tile >= NTILES) return;              // wave-uniform: EXEC stays all-1s

    const int n     = lane & 15;             // A row (M) and B column (N) index
    const int khalf = lane >> 4;             // upper lane half -> k offset +2
    const float* arow = x + (size_t)(tile * 16 + n) * HID;

    v8f acc[8] = {};
#pragma unroll 4
    for (int kk = 0; kk < 32; ++kk) {
        const int koff = kk * 4 + 2 * khalf;
        v2f a = ld2(arow + koff);
#pragma unroll
        for (int jt = 0; jt < 8; ++jt) {
            v2f b = ld2(Ww + (size_t)(jt * 16 + n) * HID + koff);
            acc[jt] = __builtin_amdgcn_wmma_f32_16x16x4_f32(
                false, a, false, b, (short)0, acc[jt], false, false);
        }
    }
#pragma unroll
    for (int jt = 0; jt < 8; ++jt) {
        const float bias = Wb[jt * 16 + n];
        float* outc = h + jt * 16 + n;
#pragma unroll
        for (int v = 0; v < 8; ++v) {
            const int m = tile * 16 + v + 8 * khalf;   // C/D layout: M = v + 8*(lane>>4)
            outc[(size_t)m * HID] = acc[jt][v] + bias;
        }
    }
}

// ---------------------------------------------------------------------------
// Kernel 2: agg[tgt[e]] += h[src[e]]  -- one wave per edge, lane covers 4 cols,
// native L2 f32 atomics via inline asm (device scope).
// ---------------------------------------------------------------------------
__global__ __launch_bounds__(256) void gnn_scatter_kernel(const float* __restrict__ h,
                                                          const int* __restrict__ src,
                                                          const int* __restrict__ tgt,
                                                          float* __restrict__ agg) {
    const int lane = threadIdx.x & 31;
    const int e    = blockIdx.x * 8 + (threadIdx.x >> 5);
    if (e >= N_EDGES) return;
    const int s = src[e];
    const int t = tgt[e];
    const float4 v = *(const float4*)(h + (size_t)s * HID + lane * 4);
    const uint64_t p = (uint64_t)(uintptr_t)(agg + (size_t)t * HID + lane * 4);
    asm volatile("global_atomic_add_f32 %0, %1, off scope:SCOPE_DEV"
                 :: "v"(p), "v"(v.x) : "memory");
    asm volatile("global_atomic_add_f32 %0, %1, off offset:4 scope:SCOPE_DEV"
                 :: "v"(p), "v"(v.y) : "memory");
    asm volatile("global_atomic_add_f32 %0, %1, off offset:8 scope:SCOPE_DEV"
                 :: "v"(p), "v"(v.z) : "memory");
    asm volatile("global_atomic_add_f32 %0, %1, off offset:12 scope:SCOPE_DEV"
                 :: "v"(p), "v"(v.w) : "memory");
}

// ---------------------------------------------------------------------------
// Kernel 3: x_new = relu(agg + u @ phi_w^T + phi_b)  (same GEMM shape as K1,
// fused with the agg add, bias and ReLU). Overwrites the h staging region.
// ---------------------------------------------------------------------------
__global__ __launch_bounds__(256) void gnn_gemm_xnew_kernel(const float* __restrict__ u,
                                                            const float* __restrict__ phw,
                                                            const float* __restrict__ phb,
                                                            const float* __restrict__ agg,
                                                            float* __restrict__ xnew) {
    const int lane  = threadIdx.x & 31;
    const int wave  = threadIdx.x >> 5;
    const int tile  = blockIdx.x * 8 + wave;
    if (tile >= NTILES) return;

    const int n     = lane & 15;
    const int khalf = lane >> 4;
    const float* arow = u + (size_t)(tile * 16 + n) * HID;

    v8f acc[8] = {};
#pragma unroll 4
    for (int kk = 0; kk < 32; ++kk) {
        const int koff = kk * 4 + 2 * khalf;
        v2f a = ld2(arow + koff);
#pragma unroll
        for (int jt = 0; jt < 8; ++jt) {
            v2f b = ld2(phw + (size_t)(jt * 16 + n) * HID + koff);
            acc[jt] = __builtin_amdgcn_wmma_f32_16x16x4_f32(
                false, a, false, b, (short)0, acc[jt], false, false);
        }
    }
#pragma unroll
    for (int jt = 0; jt < 8; ++jt) {
        const float bias = phb[jt * 16 + n];
#pragma unroll
        for (int v = 0; v < 8; ++v) {
            const int m = tile * 16 + v + 8 * khalf;
            const size_t idx = (size_t)m * HID + jt * 16 + n;
            float val = acc[jt][v] + bias + agg[idx];
            xnew[idx] = val > 0.f ? val : 0.f;
        }
    }
}

// ---------------------------------------------------------------------------
// Kernel 4: y = x_new @ pred_w^T + pred_b   ([N,128] x [128,64])
// ---------------------------------------------------------------------------
__global__ __launch_bounds__(256) void gnn_gemm_pred_kernel(const float* __restrict__ xnew,
                                                            const float* __restrict__ pw,
                                                            const float* __restrict__ pb,
                                                            float* __restrict__ y) {
    const int lane  = threadIdx.x & 31;
    const int wave  = threadIdx.x >> 5;
    const int tile  = blockIdx.x * 8 + wave;
    if (tile >= NTILES) return;

    const int n     = lane & 15;
    const int khalf = lane >> 4;
    const float* arow = xnew + (size_t)(tile * 16 + n) * HID;

    v8f acc[4] = {};
#pragma unroll 4
    for (int kk = 0; kk < 32; ++kk) {
        const int koff = kk * 4 + 2 * khalf;
        v2f a = ld2(arow + koff);
#pragma unroll
        for (int jt = 0; jt < 4; ++jt) {
            v2f b = ld2(pw + (size_t)(jt * 16 + n) * HID + koff);
            acc[jt] = __builtin_amdgcn_wmma_f32_16x16x4_f32(
                false, a, false, b, (short)0, acc[jt], false, false);
        }
    }
#pragma unroll
    for (int jt = 0; jt < 4; ++jt) {
        const float bias = pb[jt * 16 + n];
#pragma unroll
        for (int v = 0; v < 8; ++v) {
            const int m = tile * 16 + v + 8 * khalf;
            y[(size_t)m * PRED_C + jt * 16 + n] = acc[jt][v] + bias;
        }
    }
}

// ---------------------------------------------------------------------------
// Host-side launch. d_in order: x, u, edge_index, W_w, W_b, phi_w, phi_b,
// pred_w, pred_b. d_out = [x_new (N*128) | y (N*64)]. agg lives in d_ws.
// h is staged in the x_new region of d_out (dead after the scatter).
// ---------------------------------------------------------------------------
extern "C" void kernel_launch(void* const* d_in, const int* in_sizes, int n_in,
                              void* d_out, int out_size, void* d_ws, size_t ws_size,
                              hipStream_t stream) {
    const float* x   = (const float*)d_in[0];
    const float* u   = (const float*)d_in[1];
    const int*   ei  = (const int*)d_in[2];
    const float* Ww  = (const float*)d_in[3];
    const float* Wb  = (const float*)d_in[4];
    const float* phw = (const float*)d_in[5];
    const float* phb = (const float*)d_in[6];
    const float* pw  = (const float*)d_in[7];
    const float* pb  = (const float*)d_in[8];

    float* out  = (float*)d_out;
    float* xnew = out;                               // N_NODES*HID floats
    float* y    = out + (size_t)N_NODES * HID;       // N_NODES*PRED_C floats
    float* agg  = (float*)d_ws;                      // N_NODES*HID floats
    float* h    = xnew;                              // staging (overwritten by K3)

    const int n4 = N_NODES * HID / 4;                // 3,200,000 (exact /256)
    gnn_zero_kernel<<<n4 / 256, 256, 0, stream>>>((float4*)agg, n4);

    const int gemm_blocks = (NTILES + 7) / 8;        // 782
    gnn_gemm_h_kernel<<<gemm_blocks, 256, 0, stream>>>(x, Ww, Wb, h);

    gnn_scatter_kernel<<<N_EDGES / 8, 256, 0, stream>>>(h, ei, ei + N_EDGES, agg);

    gnn_gemm_xnew_kernel<<<gemm_blocks, 256, 0, stream>>>(u, phw, phb, agg, xnew);

    gnn_gemm_pred_kernel<<<gemm_blocks, 256, 0, stream>>>(xnew, pw, pb, y);
}